// GeoFeaturizer_74629351735695
// MI455X (gfx1250) — compile-verified
//
#include <hip/hip_runtime.h>
#include <hip/hip_bf16.h>
#include <math.h>

#define N_NODES 30000
#define N_EDGES 600000
#define PER     3750      // nodes per batch
#define NVROWS  780000    // E + 2*K*N

// ---- flat float offsets into d_out (return-order concat) ----
constexpr long long OFF_V        = 0LL;
constexpr long long OFF_E        = 2281824LL;     // (N+24)*76
constexpr long long OFF_TROT     = 168421824LL;   // + 780000*213
constexpr long long OFF_TTRANS   = 168692040LL;
constexpr long long OFF_TTSROT   = 168782112LL;
constexpr long long OFF_TTSTRANS = 175802112LL;
constexpr long long OFF_BATCH    = 178142112LL;
constexpr long long OFF_EIDX     = 178172136LL;
constexpr long long OFF_CHAIN    = 179732136LL;

typedef float v2f __attribute__((ext_vector_type(2)));
typedef float v4f __attribute__((ext_vector_type(4)));
typedef float v8f __attribute__((ext_vector_type(8)));

// rbf over 16 centers mu_k = 20k/15, sigma = 1.25  (fast exp: arg <= 0, no range issue)
__device__ __forceinline__ void rbf16(float* dst, float nrm) {
#pragma unroll
  for (int k = 0; k < 16; ++k) {
    float tt = (nrm - 1.3333333333f * (float)k) * 0.8f;
    dst[k] = __expf(-tt * tt);
  }
}

// decouple() for a 3-vector -> 19 floats (3 dir + 16 rbf)
__device__ __forceinline__ void decouple3(float* dst, float u0, float u1, float u2) {
  float nrm = sqrtf(u0 * u0 + u1 * u1 + u2 * u2 + 1e-12f);
  float inv = 1.0f / (nrm + 1e-6f);
  bool small = nrm < 1e-4f;
  dst[0] = small ? 0.f : u0 * inv;
  dst[1] = small ? 0.f : u1 * inv;
  dst[2] = small ? 0.f : u2 * inv;
  rbf16(dst + 3, nrm);
}

// ---------------- per-node: frames R,t + V features ----------------
__global__ void node_kernel(const float* __restrict__ X, const int* __restrict__ batch_id,
                            const int* __restrict__ chain, float* __restrict__ out,
                            float* __restrict__ Rws, float* __restrict__ tws) {
  int n = blockIdx.x * blockDim.x + threadIdx.x;
  if (n >= N_NODES) return;
  const float* x = X + (long long)n * 12;
  float nx = x[0] - x[3], nyv = x[1] - x[4], nz = x[2] - x[5];
  float cx = x[6] - x[3], cy  = x[7] - x[4], cz = x[8] - x[5];
  float nrm  = sqrtf(1e-20f + cx * cx + cy * cy);
  float s1 = -cy / nrm, c1 = cx / nrm;
  float nrm2 = sqrtf(1e-20f + cx * cx + cy * cy + cz * cz);
  float s2 = cz / nrm2, c2 = sqrtf(cx * cx + cy * cy) / nrm2;
  // Rc = R2*R1
  float Rc00 = c2 * c1, Rc01 = -c2 * s1, Rc02 = s2;
  float Rc10 = s1,      Rc11 = c1,       Rc12 = 0.f;
  float Rc20 = -s2 * c1, Rc21 = s2 * s1, Rc22 = c2;
  float nr1 = Rc10 * nx + Rc11 * nyv + Rc12 * nz;
  float nr2 = Rc20 * nx + Rc21 * nyv + Rc22 * nz;
  float nrm3 = sqrtf(1e-20f + nr1 * nr1 + nr2 * nr2);
  float sn = -nr2 / nrm3, cn = nr1 / nrm3;
  // M = Rn*Rc ; R = M^T (this is what the reference returns)
  float M00 = Rc00, M01 = Rc01, M02 = Rc02;
  float M10 = cn * Rc10 - sn * Rc20, M11 = cn * Rc11 - sn * Rc21, M12 = cn * Rc12 - sn * Rc22;
  float M20 = sn * Rc10 + cn * Rc20, M21 = sn * Rc11 + cn * Rc21, M22 = sn * Rc12 + cn * Rc22;
  float R[9];
  R[0] = M00; R[1] = M10; R[2] = M20;
  R[3] = M01; R[4] = M11; R[5] = M21;
  R[6] = M02; R[7] = M12; R[8] = M22;
  float t0 = x[3], t1 = x[4], t2 = x[5];
#pragma unroll
  for (int i = 0; i < 9; ++i) {
    Rws[n * 9 + i] = R[i];
    out[OFF_TROT + (long long)n * 9 + i] = R[i];
  }
  tws[n * 3 + 0] = t0; tws[n * 3 + 1] = t1; tws[n * 3 + 2] = t2;
  out[OFF_TTRANS + (long long)n * 3 + 0] = t0;
  out[OFF_TTRANS + (long long)n * 3 + 1] = t1;
  out[OFF_TTRANS + (long long)n * 3 + 2] = t2;
  out[OFF_BATCH + n] = (float)batch_id[n];
  out[OFF_CHAIN + n] = (float)chain[n];
  // backbone deltas -> local frame -> decouple -> V row (76 floats)
  bool z = (n % PER) == 0;   // batch starts get zeroed d
  float* vrow = out + OFF_V + (long long)n * 76;
#pragma unroll
  for (int a = 0; a < 4; ++a) {
    float d0, d1, d2;
    if (z) { d0 = d1 = d2 = 0.f; }
    else {
      const float* cur = X + ((long long)n * 4 + a) * 3;
      d0 = cur[0] - cur[-3]; d1 = cur[1] - cur[-2]; d2 = cur[2] - cur[-1];
    }
    float l0 = R[0] * d0 + R[3] * d1 + R[6] * d2;  // R^T . d (einsum nji,naj->nai)
    float l1 = R[1] * d0 + R[4] * d1 + R[7] * d2;
    float l2 = R[2] * d0 + R[5] * d1 + R[8] * d2;
    decouple3(vrow + a * 19, l0, l1, l2);
  }
}

// ---------------- per-edge: 213-float Efeat rows via LDS + WMMA pos ----------------
// LDS: 64x213 Efeat rows | 64x9 Rts rows | 64x3 tts rows = 14,400 floats = 57.6 KB
__global__ __launch_bounds__(64) void edge_kernel(const float* __restrict__ X,
                                                  const int* __restrict__ node_idx,
                                                  const int* __restrict__ eidx,
                                                  const float* __restrict__ Rws,
                                                  const float* __restrict__ tws,
                                                  float* __restrict__ out) {
  __shared__ __align__(16) float srow[64 * 213 + 64 * 9 + 64 * 3];
  float* sR = srow + 64 * 213;   // byte offset 54,528 (16B aligned)
  float* sT = sR + 64 * 9;       // byte offset 56,832 (16B aligned)
  const int tid = threadIdx.x;
  const int lane = tid & 31;
  const int wave = tid >> 5;
  const int e = blockIdx.x * 64 + tid;            // E divisible by 64 -> full waves, EXEC all-1s
  const int e0 = blockIdx.x * 64 + wave * 32;     // wave's first edge
  float* row = &srow[tid * 213];

  const int src = eidx[e];
  const int dst = eidx[N_EDGES + e];
  float Rs[9], Rd[9];
#pragma unroll
  for (int i = 0; i < 9; ++i) { Rs[i] = Rws[src * 9 + i]; Rd[i] = Rws[dst * 9 + i]; }
  float ts0 = tws[src * 3], ts1 = tws[src * 3 + 1], ts2 = tws[src * 3 + 2];
  float td0 = tws[dst * 3], td1 = tws[dst * 3 + 1], td2 = tws[dst * 3 + 2];

  // diffE: 8 points (4 src atoms, 4 dst atoms) in src frame -> cols 0..151
#pragma unroll
  for (int a = 0; a < 8; ++a) {
    const float* p = (a < 4) ? (X + ((long long)src * 4 + a) * 3)
                             : (X + ((long long)dst * 4 + (a - 4)) * 3);
    float q0 = p[0] - ts0, q1 = p[1] - ts1, q2 = p[2] - ts2;
    float l0 = Rs[0] * q0 + Rs[3] * q1 + Rs[6] * q2;
    float l1 = Rs[1] * q0 + Rs[4] * q1 + Rs[7] * q2;
    float l2 = Rs[2] * q0 + Rs[5] * q1 + Rs[8] * q2;
    decouple3(row + a * 19, l0, l1, l2);
  }
  // Rts = Rd^T * Rs   (einsum eji,ejk->eik)
  float Rts[9];
#pragma unroll
  for (int i = 0; i < 3; ++i)
#pragma unroll
    for (int k = 0; k < 3; ++k)
      Rts[i * 3 + k] = Rd[0 + i] * Rs[0 + k] + Rd[3 + i] * Rs[3 + k] + Rd[6 + i] * Rs[6 + k];
  // E_quant = Rts^T flattened -> cols 152..160; stage Rts row for coalesced writeout
#pragma unroll
  for (int i = 0; i < 3; ++i)
#pragma unroll
    for (int k = 0; k < 3; ++k)
      row[152 + i * 3 + k] = Rts[k * 3 + i];
#pragma unroll
  for (int m = 0; m < 9; ++m) sR[tid * 9 + m] = Rts[m];
  // tts = Rd^T * (t_src - t_dst) -> decouple -> cols 161..179
  float u0 = ts0 - td0, u1 = ts1 - td1, u2 = ts2 - td2;
  float w0 = Rd[0] * u0 + Rd[3] * u1 + Rd[6] * u2;
  float w1 = Rd[1] * u0 + Rd[4] * u1 + Rd[7] * u2;
  float w2 = Rd[2] * u0 + Rd[5] * u1 + Rd[8] * u2;
  decouple3(row + 161, w0, w1, w2);
  sT[tid * 3 + 0] = w0; sT[tid * 3 + 1] = w1; sT[tid * 3 + 2] = w2;
  // E_bias: 1 dir + 16 rbf -> cols 196..212
  float bb = (float)(node_idx[src] - node_idx[dst]);
  float nb = sqrtf(bb * bb + 1e-12f);
  row[196] = (nb < 1e-4f) ? 0.f : bb / (nb + 1e-6f);
  rbf16(row + 197, nb);

  // ---- pos (int_embedding, 16 cols @180): rank-1 outer product via WMMA f32 16x16x4 ----
  float dval = (float)(src - dst);
  float dhi  = __shfl(dval, (lane & 15) + 16, 32);  // upper-half edges for 2nd WMMA
  v2f a0, a1, b0;
  a0.x = (lane < 16) ? dval : 0.f; a0.y = 0.f;      // A[m][0] = delta_m, other K rows 0
  a1.x = (lane < 16) ? dhi  : 0.f; a1.y = 0.f;
  b0.x = (lane < 8) ? __expf(-(float)lane * 1.1512925465f) : 0.f;  // freq_j = 10^(-j/2)
  b0.y = 0.f;                                        // B[0][n] = freq_n, other rows 0
  v8f c0 = {0.f, 0.f, 0.f, 0.f, 0.f, 0.f, 0.f, 0.f};
#if __has_builtin(__builtin_amdgcn_wmma_f32_16x16x4_f32)
  v8f dlo = __builtin_amdgcn_wmma_f32_16x16x4_f32(false, a0, false, b0, (short)0, c0, false, false);
  v8f dup = __builtin_amdgcn_wmma_f32_16x16x4_f32(false, a1, false, b0, (short)0, c0, false, false);
  {
    int nn = lane & 15;
    if (nn < 8) {
      int mh = (lane >= 16) ? 8 : 0;
      float* wbase = &srow[wave * 32 * 213];
#pragma unroll
      for (int r = 0; r < 8; ++r) {
        float sv, cv, sv2, cv2;
        __sincosf(dlo[r], &sv, &cv);                  // D: M = r+mh, N = nn
        __sincosf(dup[r], &sv2, &cv2);
        float* r0 = wbase + (r + mh) * 213;
        float* r1 = wbase + (r + mh + 16) * 213;
        r0[180 + nn] = cv; r0[188 + nn] = sv;
        r1[180 + nn] = cv2; r1[188 + nn] = sv2;
      }
    }
  }
#else
#pragma unroll
  for (int j = 0; j < 8; ++j) {
    float ang = dval * __expf(-(float)j * 1.1512925465f);
    float sv, cv; __sincosf(ang, &sv, &cv);
    row[180 + j] = cv; row[188 + j] = sv;
  }
#endif

  // eidx head: contiguous per-lane stores (coalesced)
  out[OFF_EIDX + e] = (float)src;
  out[OFF_EIDX + NVROWS + e] = (float)dst;

  __syncthreads();
  // coalesced non-temporal b128 writeout: Efeat 32 rows = 6816 floats = 1704 x v4f
  {
    const v4f* s4 = (const v4f*)&srow[wave * 6816];
    v4f* g4 = (v4f*)(out + OFF_E + (long long)e0 * 213LL);
    for (int i = lane; i < 1704; i += 32) __builtin_nontemporal_store(s4[i], &g4[i]);
  }
  // Tts_rot: 32 rows x 9 = 288 floats = 72 x v4f
  {
    const v4f* r4 = (const v4f*)&sR[wave * 288];
    v4f* gr4 = (v4f*)(out + OFF_TTSROT + (long long)e0 * 9LL);
#pragma unroll
    for (int i = lane; i < 72; i += 32) __builtin_nontemporal_store(r4[i], &gr4[i]);
  }
  // Tts_trans: 32 rows x 3 = 96 floats = 24 x v4f
  {
    const v4f* t4 = (const v4f*)&sT[wave * 96];
    v4f* gt4 = (v4f*)(out + OFF_TTSTRANS + (long long)e0 * 3LL);
    if (lane < 24) __builtin_nontemporal_store(t4[lane], &gt4[lane]);
  }
}

// ---------------- tails ----------------
__global__ void zero4_kernel(v4f* __restrict__ p, int n4) {
  int i = blockIdx.x * blockDim.x + threadIdx.x;
  int stride = gridDim.x * blockDim.x;
  v4f z = {0.f, 0.f, 0.f, 0.f};
  for (; i < n4; i += stride) __builtin_nontemporal_store(z, &p[i]);
}

__global__ void eye_tail_kernel(float* __restrict__ out) {
  long long i = (long long)blockIdx.x * 256 + threadIdx.x;
  if (i >= 1620000LL) return;   // 180000 identity 3x3 for Tts_rot tail
  int c = (int)(i % 9);
  float v = (c == 0 || c == 4 || c == 8) ? 1.f : 0.f;
  __builtin_nontemporal_store(v, &out[OFF_TTSROT + 5400000LL + i]);
}

__global__ void eidx_tail_kernel(float* __restrict__ out) {
  int j = blockIdx.x * 256 + threadIdx.x;
  if (j >= 180000) return;
  int jm = (j < 90000) ? j : j - 90000;
  int nnode = jm % N_NODES;
  int kk = jm / N_NODES;
  float gs = (float)(N_NODES + nnode / PER + kk * 8);  // N + batch_id + k*B
  float gd = (float)nnode;
  bool first = j < 90000;
  __builtin_nontemporal_store(first ? gs : gd, &out[OFF_EIDX + N_EDGES + j]);
  __builtin_nontemporal_store(first ? gd : gs, &out[OFF_EIDX + NVROWS + N_EDGES + j]);
}

__global__ void misc_kernel(float* __restrict__ out) {
  int t = threadIdx.x;
  // V_g: 24 rows x 76 (int_embedding of repeat(arange(3),8), num_emb=76)
  for (int idx = t; idx < 24 * 76; idx += 256) {
    int r = idx / 76, c = idx % 76;
    float dv = (float)(r / 8);
    int j = (c < 38) ? c : c - 38;
    float fr = __expf(-(float)(2 * j) * 9.210340371976184f / 76.f);
    float ang = dv * fr;
    float sv, cv; __sincosf(ang, &sv, &cv);
    out[OFF_V + (long long)(N_NODES + r) * 76 + c] = (c < 38) ? cv : sv;
  }
  // T_rot tail: 24 identity 3x3
  for (int idx = t; idx < 216; idx += 256) {
    int c = idx % 9;
    out[OFF_TROT + 270000LL + idx] = (c == 0 || c == 4 || c == 8) ? 1.f : 0.f;
  }
  // T_trans tail zeros
  for (int idx = t; idx < 72; idx += 256) out[OFF_TTRANS + 90000LL + idx] = 0.f;
  // batch/chain tails
  if (t < 24) {
    out[OFF_BATCH + N_NODES + t] = (float)(t % 8);   // tile(arange(8),3)
    out[OFF_CHAIN + N_NODES + t] = 1001.f;
  }
}

extern "C" void kernel_launch(void* const* d_in, const int* in_sizes, int n_in,
                              void* d_out, int out_size, void* d_ws, size_t ws_size,
                              hipStream_t stream) {
  const float* X        = (const float*)d_in[0];
  const int* node_idx   = (const int*)d_in[1];
  const int* eidx       = (const int*)d_in[2];
  const int* batch_id   = (const int*)d_in[3];
  const int* chain      = (const int*)d_in[4];
  float* out = (float*)d_out;
  float* Rws = (float*)d_ws;           // 270,000 floats
  float* tws = Rws + 270000;           // 90,000 floats (1.44 MB total ws)

  node_kernel<<<(N_NODES + 255) / 256, 256, 0, stream>>>(X, batch_id, chain, out, Rws, tws);
  zero4_kernel<<<2048, 256, 0, stream>>>((v4f*)(out + 130081824LL), 9585000);  // E_g zeros
  zero4_kernel<<<512, 256, 0, stream>>>((v4f*)(out + 177602112LL), 135000);    // Tts_trans tail
  eye_tail_kernel<<<(1620000 + 255) / 256, 256, 0, stream>>>(out);
  eidx_tail_kernel<<<(180000 + 255) / 256, 256, 0, stream>>>(out);
  misc_kernel<<<1, 256, 0, stream>>>(out);
  edge_kernel<<<N_EDGES / 64, 64, 0, stream>>>(X, node_idx, eidx, Rws, tws, out);
}